// MSGGRUCell_78108275245104
// MI455X (gfx1250) — compile-verified
//
#include <hip/hip_runtime.h>
#include <math.h>

// Problem dims (match reference)
#define NN 1024          // nodes
#define BB 32            // batch
#define UU 64            // units
#define FF 66            // input feature dim inside gconv
#define FB (FF * BB)     // 2112 columns of x0

typedef __attribute__((ext_vector_type(2))) float v2f;
typedef __attribute__((ext_vector_type(8))) float v8f;

#define GLOBAL_AS __attribute__((address_space(1)))
#define LDS_AS    __attribute__((address_space(3)))

#if defined(__has_builtin)
#if __has_builtin(__builtin_amdgcn_global_load_async_to_lds_b32) && \
    __has_builtin(__builtin_amdgcn_s_wait_asynccnt)
#define HAVE_ASYNC_LDS 1
#endif
#endif
#ifndef HAVE_ASYNC_LDS
#define HAVE_ASYNC_LDS 0
#endif

// Interleaved K-pair LDS layout: element (k, x) lives at
//   ((k/2)*LD + x)*2 + (k&1)
// so the two K values a lane needs sit in one aligned 8-byte slot
// -> single ds_load_b64 into an even VGPR pair (WMMA operand ready).
__device__ __forceinline__ int ldsidx(int k, int x, int LD) {
  return (((k >> 1) * LD + x) << 1) | (k & 1);
}

// ---------------------------------------------------------------------------
// WMMA micro-kernel: one 16x16 f32 tile over a K-chunk of 16 (4 k=4 steps).
// A frag (16x4 f32): lanes 0-15 hold K={0,1}, lanes 16-31 hold K={2,3}.
// B frag (4x16 f32): same split, N striped across lanes.
// C/D: 8 VGPRs, VGPR r = row r (lanes 0-15) / row 8+r (lanes 16-31).
// ---------------------------------------------------------------------------
__device__ __forceinline__ v8f wmma16(const float* __restrict__ As,
                                      const float* __restrict__ Bs,
                                      int LDA, int LDB, int m0, int n0,
                                      int lane, v8f acc) {
  const int half = lane >> 4;
  const int l = lane & 15;
#pragma unroll
  for (int ks = 0; ks < 4; ++ks) {
    const int kp = ks * 2 + half;  // K-pair index = (4*ks + 2*half)/2
    v2f a = *(const v2f*)&As[(kp * LDA + m0 + l) << 1];
    v2f b = *(const v2f*)&Bs[(kp * LDB + n0 + l) << 1];
    acc = __builtin_amdgcn_wmma_f32_16x16x4_f32(false, a, false, b, (short)0,
                                                acc, false, false);
  }
  return acc;
}

__device__ __forceinline__ void store16(float* __restrict__ dst, long ldd,
                                        int row0, int col0, v8f acc, int lane) {
  const int half = lane >> 4;
  const int l = lane & 15;
#pragma unroll
  for (int r = 0; r < 8; ++r)
    dst[(long)(row0 + r + 8 * half) * ldd + col0 + l] = acc[r];
}

// ---------------------------------------------------------------------------
// K1: build x0[n, f*B + b] = concat(inputs, state) transposed to (N, F*B)
// ---------------------------------------------------------------------------
__global__ void build_x0_kernel(const float* __restrict__ inp,
                                const float* __restrict__ state,
                                float* __restrict__ x0) {
  int idx = blockIdx.x * blockDim.x + threadIdx.x;
  const int tot = NN * FB;
  if (idx >= tot) return;
  int n = idx / FB;
  int rem = idx - n * FB;
  int f = rem / BB;
  int b = rem - f * BB;
  float v = (f < 2) ? inp[b * (NN * 2) + n * 2 + f]
                    : state[b * (NN * UU) + n * UU + (f - 2)];
  x0[idx] = v;
}

// ---------------------------------------------------------------------------
// K2 (dominant): xk[m] = mats[m] @ x0.
// A generated on the fly from adj & mask (both L2-resident), register-
// prefetch pipelined.  B (x0) tile double-buffered in LDS via
// GLOBAL_LOAD_ASYNC_TO_LDS (ASYNCcnt) when available.
// grid: (FB/64, NN/64, 10), block 256 (8 waves -> 4x2 grid, 16x32 each)
// ---------------------------------------------------------------------------
__global__ __launch_bounds__(256) void support_gemm_kernel(
    const float* __restrict__ adj, const float* __restrict__ mask,
    const float* __restrict__ x0, float* __restrict__ xk) {
  __shared__ __align__(16) float As[16 * 64];
  __shared__ __align__(16) float Bs[2][16 * 64];
  const int m = blockIdx.z;
  const int i0 = blockIdx.y * 64;
  const int c0 = blockIdx.x * 64;
  const int tid = threadIdx.x, lane = tid & 31, w = tid >> 5;
  const int wr = w & 3, wc = w >> 2;
  const int sel = (m >= 5) ? 1 : 0;
  const bool hasMask = (m >= 1 && m <= 8);
  const float* A = adj + (size_t)sel * NN * NN;
  const float* Mk = mask + (size_t)(m - 1) * NN * NN;

  float aReg[4];

  auto loadA = [&](int j0) {
    if (j0 + 32 < NN)  // keep the next-next A rows warm (global_prefetch_b8)
      __builtin_prefetch(&A[(size_t)(i0 + (tid >> 4)) * NN + j0 + 32], 0, 3);
#pragma unroll
    for (int e = 0; e < 4; ++e) {
      int p = tid + e * 256;
      int kk = p & 15, mm = p >> 4;
      size_t g = (size_t)(i0 + mm) * NN + (j0 + kk);
      float v = A[g];
      if (hasMask) v *= Mk[g];
      aReg[e] = v;
    }
  };
  auto storeA = [&]() {
#pragma unroll
    for (int e = 0; e < 4; ++e) {
      int p = tid + e * 256;
      As[ldsidx(p & 15, p >> 4, 64)] = aReg[e];
    }
  };
  auto issueB = [&](int j0, int buf) {
#pragma unroll
    for (int e = 0; e < 4; ++e) {
      int p = tid + e * 256;
      int kb = p >> 6, cc = p & 63;
      const float* g = x0 + (size_t)(j0 + kb) * FB + c0 + cc;
      float* l = &Bs[buf][ldsidx(kb, cc, 64)];
#if HAVE_ASYNC_LDS
      __builtin_amdgcn_global_load_async_to_lds_b32(
          (GLOBAL_AS int*)g, (LDS_AS int*)l, 0, 0);
#else
      *l = *g;
#endif
    }
  };

  v8f acc0 = {}, acc1 = {};
  loadA(0);
  issueB(0, 0);
  const int NIT = NN / 16;  // 64 K-chunks
  for (int it = 0; it < NIT; ++it) {
    const int cur = it & 1;
    storeA();  // A tile for this chunk (regs loaded last iteration)
    const bool more = (it + 1 < NIT);
    if (more) issueB((it + 1) * 16, 1 - cur);  // overlap next B with compute
#if HAVE_ASYNC_LDS
    if (more) __builtin_amdgcn_s_wait_asynccnt(4);  // current buf landed
    else      __builtin_amdgcn_s_wait_asynccnt(0);
#endif
    __syncthreads();
    if (more) loadA((it + 1) * 16);  // next A globals overlap the WMMAs
    acc0 = wmma16(As, Bs[cur], 64, 64, wr * 16, wc * 32, lane, acc0);
    acc1 = wmma16(As, Bs[cur], 64, 64, wr * 16, wc * 32 + 16, lane, acc1);
    __syncthreads();
  }
  float* dst = xk + (size_t)m * NN * FB;
  store16(dst, FB, i0 + wr * 16, c0 + wc * 32, acc0, lane);
  store16(dst, FB, i0 + wr * 16, c0 + wc * 32 + 16, acc1, lane);
}

// ---------------------------------------------------------------------------
// K3: projection  xp[k][b][n][o] = sum_f xall[k][n, f*B + b] * W[f][o]
// grid: (O/64, NN/64, 11*BB); K padded 66 -> 80 (5 chunks of 16)
// ---------------------------------------------------------------------------
__global__ __launch_bounds__(256) void proj_gemm_kernel(
    const float* __restrict__ x0, const float* __restrict__ xk,
    const float* __restrict__ W, float* __restrict__ xp, int O) {
  __shared__ __align__(16) float As[16 * 64];
  __shared__ __align__(16) float Bs[16 * 64];
  const int z = blockIdx.z;
  const int k = z / BB;
  const int b = z - k * BB;
  const int n0 = blockIdx.y * 64;
  const int o0 = blockIdx.x * 64;
  const int tid = threadIdx.x, lane = tid & 31, w = tid >> 5;
  const int wr = w & 3, wc = w >> 2;
  const float* src = (k == 0) ? x0 : (xk + (size_t)(k - 1) * NN * FB);

  v8f acc0 = {}, acc1 = {};
  for (int kc = 0; kc < 5; ++kc) {
    __syncthreads();
#pragma unroll
    for (int e = 0; e < 4; ++e) {
      int p = tid + e * 256;
      int kk = p & 15, mm = p >> 4;
      int f = kc * 16 + kk;
      As[ldsidx(kk, mm, 64)] =
          (f < FF) ? src[(size_t)(n0 + mm) * FB + f * BB + b] : 0.f;
      int cc = p & 63, kb = p >> 6;
      int f2 = kc * 16 + kb;
      Bs[ldsidx(kb, cc, 64)] = (f2 < FF) ? W[(size_t)f2 * O + o0 + cc] : 0.f;
    }
    __syncthreads();
    acc0 = wmma16(As, Bs, 64, 64, wr * 16, wc * 32, lane, acc0);
    acc1 = wmma16(As, Bs, 64, 64, wr * 16, wc * 32 + 16, lane, acc1);
  }
  float* dst = xp + ((size_t)k * BB + b) * NN * O;
  store16(dst, O, n0 + wr * 16, o0 + wc * 32, acc0, lane);
  store16(dst, O, n0 + wr * 16, o0 + wc * 32 + 16, acc1, lane);
}

// ---------------------------------------------------------------------------
// K4: per-node attention.  One block per node n, 320 threads (10 waves).
// h = relu([160 x O] @ Watt[n]); s = h . Watt1[n]; softmax over 5 supports;
// dst[b][n][o] = sum_k a[b,k] * xp[kbase+k][b][n][o]
// ---------------------------------------------------------------------------
__global__ __launch_bounds__(320) void attention_kernel(
    const float* __restrict__ xp, const float* __restrict__ Watt,
    const float* __restrict__ Watt1, float* __restrict__ dst, int O,
    int kbase) {
  __shared__ __align__(16) float As[16 * 160];
  __shared__ __align__(16) float Bs[16 * 128];
  __shared__ float s_sh[160];
  __shared__ float a_sh[160];
  const int n = blockIdx.x;
  const int tid = threadIdx.x, lane = tid & 31, w = tid >> 5;  // w in 0..9
  const int ncol = O >> 4;  // output col tiles (8 or 4)
  const int nkc = O >> 4;   // K chunks of 16

  if (tid < 160) s_sh[tid] = 0.f;

  v8f zero = {};
  v8f acc[8];
#pragma unroll
  for (int t = 0; t < 8; ++t) acc[t] = zero;

  const float* WB = Watt + (size_t)n * O * O;
  for (int kc = 0; kc < nkc; ++kc) {
    __syncthreads();
    for (int p = tid; p < 160 * 16; p += 320) {
      int row = p % 160, kk = p / 160;
      int b = row / 5, kk2 = row - b * 5;
      As[ldsidx(kk, row, 160)] =
          xp[(((size_t)(kbase + kk2) * BB + b) * NN + n) * O + kc * 16 + kk];
    }
    for (int p = tid; p < 16 * O; p += 320) {
      int c = p % O, kk = p / O;
      Bs[ldsidx(kk, c, O)] = WB[(size_t)(kc * 16 + kk) * O + c];
    }
    __syncthreads();
    for (int ct = 0; ct < ncol; ++ct)
      acc[ct] = wmma16(As, Bs, 160, O, w * 16, ct * 16, lane, acc[ct]);
  }

  // relu(h) . Watt1 -> per-row score (LDS atomic reduction across lanes)
  float part[8];
#pragma unroll
  for (int r = 0; r < 8; ++r) part[r] = 0.f;
  for (int ct = 0; ct < ncol; ++ct) {
    int col = ct * 16 + (lane & 15);
    float wv = Watt1[(size_t)n * O + col];
#pragma unroll
    for (int r = 0; r < 8; ++r) {
      float h = acc[ct][r];
      h = h > 0.f ? h : 0.f;
      part[r] += h * wv;
    }
  }
  const int half = lane >> 4;
#pragma unroll
  for (int r = 0; r < 8; ++r)
    atomicAdd(&s_sh[w * 16 + r + 8 * half], part[r]);
  __syncthreads();

  // softmax over the 5 supports, one thread per batch element
  if (tid < BB) {
    int b = tid;
    float sv[5], mx = -1e30f;
#pragma unroll
    for (int kk = 0; kk < 5; ++kk) {
      sv[kk] = s_sh[b * 5 + kk];
      mx = fmaxf(mx, sv[kk]);
    }
    float sum = 0.f;
#pragma unroll
    for (int kk = 0; kk < 5; ++kk) {
      sv[kk] = expf(sv[kk] - mx);
      sum += sv[kk];
    }
#pragma unroll
    for (int kk = 0; kk < 5; ++kk) a_sh[b * 5 + kk] = sv[kk] / sum;
  }
  __syncthreads();

  // weighted combine
  for (int idx = tid; idx < BB * O; idx += 320) {
    int b = idx / O, o = idx - b * O;
    float v = 0.f;
#pragma unroll
    for (int kk = 0; kk < 5; ++kk)
      v += a_sh[b * 5 + kk] *
           xp[(((size_t)(kbase + kk) * BB + b) * NN + n) * O + o];
    dst[((size_t)b * NN + n) * O + o] = v;
  }
}

// ---------------------------------------------------------------------------
// K5: out = act([xi|xs|xbp] @ Wout + bias); rows = B*N, K = 3*O
// act: 1=sigmoid (gate), 2=tanh (candidate)
// ---------------------------------------------------------------------------
__global__ __launch_bounds__(256) void out_gemm_kernel(
    const float* __restrict__ xi, const float* __restrict__ xs,
    const float* __restrict__ xbp, const float* __restrict__ Wout,
    const float* __restrict__ bias, float* __restrict__ dst, int O, int act) {
  __shared__ __align__(16) float As[16 * 64];
  __shared__ __align__(16) float Bs[16 * 64];
  const int r0 = blockIdx.y * 64;
  const int o0 = blockIdx.x * 64;
  const int tid = threadIdx.x, lane = tid & 31, w = tid >> 5;
  const int wr = w & 3, wc = w >> 2;
  const int nkc = (3 * O) / 16;

  v8f acc0 = {}, acc1 = {};
  for (int kc = 0; kc < nkc; ++kc) {
    __syncthreads();
#pragma unroll
    for (int e = 0; e < 4; ++e) {
      int p = tid + e * 256;
      int kk = p & 15, mm = p >> 4;
      int q = kc * 16 + kk;
      const float* srcb = (q < O) ? xi : ((q < 2 * O) ? xs : xbp);
      As[ldsidx(kk, mm, 64)] = srcb[(size_t)(r0 + mm) * O + (q % O)];
      int cc = p & 63, kb = p >> 6;
      Bs[ldsidx(kb, cc, 64)] = Wout[(size_t)(kc * 16 + kb) * O + o0 + cc];
    }
    __syncthreads();
    acc0 = wmma16(As, Bs, 64, 64, wr * 16, wc * 32, lane, acc0);
    acc1 = wmma16(As, Bs, 64, 64, wr * 16, wc * 32 + 16, lane, acc1);
  }
  const int half = lane >> 4, l = lane & 15;
#pragma unroll
  for (int t = 0; t < 2; ++t) {
    v8f a = t ? acc1 : acc0;
    int col = o0 + wc * 32 + t * 16 + l;
    float bz = bias[col];
#pragma unroll
    for (int rr = 0; rr < 8; ++rr) {
      float v = a[rr] + bz;
      v = (act == 1) ? (1.f / (1.f + expf(-v))) : tanhf(v);
      dst[(size_t)(r0 + wr * 16 + rr + 8 * half) * O + col] = v;
    }
  }
}

// ---------------------------------------------------------------------------
// K6: r*hx  (r = gate_value[..., :U])
// ---------------------------------------------------------------------------
__global__ void rhx_kernel(const float* __restrict__ gval,
                           const float* __restrict__ hx,
                           float* __restrict__ rhx) {
  int idx = blockIdx.x * blockDim.x + threadIdx.x;
  if (idx >= BB * NN * UU) return;
  int o = idx % UU;
  int bn = idx / UU;
  float r = gval[(size_t)bn * (2 * UU) + o];
  rhx[idx] = r * hx[idx];
}

// ---------------------------------------------------------------------------
// K7: out = u*hx + (1-u)*c  (u = gate_value[..., U:])
// ---------------------------------------------------------------------------
__global__ void final_kernel(const float* __restrict__ gval,
                             const float* __restrict__ hx,
                             const float* __restrict__ cbuf,
                             float* __restrict__ out) {
  int idx = blockIdx.x * blockDim.x + threadIdx.x;
  if (idx >= BB * NN * UU) return;
  int o = idx % UU;
  int bn = idx / UU;
  float u = gval[(size_t)bn * (2 * UU) + UU + o];
  out[idx] = u * hx[idx] + (1.f - u) * cbuf[idx];
}

// ---------------------------------------------------------------------------
extern "C" void kernel_launch(void* const* d_in, const int* in_sizes, int n_in,
                              void* d_out, int out_size, void* d_ws,
                              size_t ws_size, hipStream_t stream) {
  const float* inputs  = (const float*)d_in[0];
  const float* hx      = (const float*)d_in[1];
  const float* adj     = (const float*)d_in[2];
  const float* mask    = (const float*)d_in[3];
  const float* W_ru    = (const float*)d_in[4];
  const float* Watt_ru = (const float*)d_in[5];
  const float* Wt1_ru  = (const float*)d_in[6];
  const float* Wout_ru = (const float*)d_in[7];
  const float* b_ru    = (const float*)d_in[8];
  const float* W_c     = (const float*)d_in[9];
  const float* Watt_c  = (const float*)d_in[10];
  const float* Wt1_c   = (const float*)d_in[11];
  const float* Wout_c  = (const float*)d_in[12];
  const float* b_c     = (const float*)d_in[13];
  float* out = (float*)d_out;

  // workspace carve (floats); gate-sized buffers reused by candidate pass
  float* ws = (float*)d_ws;
  size_t off = 0;
  float* x0 = ws + off; off += (size_t)NN * FB;            // 8.6 MB
  float* xk = ws + off; off += (size_t)10 * NN * FB;       // 86.5 MB
  float* xp = ws + off; off += (size_t)11 * BB * NN * 128; // 184.5 MB
  float* xi = ws + off; off += (size_t)BB * NN * 128;
  float* xs = ws + off; off += (size_t)BB * NN * 128;
  float* gv = ws + off; off += (size_t)BB * NN * 128;      // gate sigmoid out
  float* rh = ws + off; off += (size_t)BB * NN * UU;       // r*hx
  float* cb = ws + off; off += (size_t)BB * NN * UU;       // candidate tanh out

  const int totx0 = NN * FB;
  const int totu = BB * NN * UU;

  // ---- gate gconv (O = 2U = 128), sigmoid ----
  build_x0_kernel<<<(totx0 + 255) / 256, 256, 0, stream>>>(inputs, hx, x0);
  support_gemm_kernel<<<dim3(FB / 64, NN / 64, 10), 256, 0, stream>>>(
      adj, mask, x0, xk);
  proj_gemm_kernel<<<dim3(2, NN / 64, 11 * BB), 256, 0, stream>>>(
      x0, xk, W_ru, xp, 128);
  attention_kernel<<<dim3(NN), 320, 0, stream>>>(xp, Watt_ru, Wt1_ru, xi, 128, 1);
  attention_kernel<<<dim3(NN), 320, 0, stream>>>(xp, Watt_ru, Wt1_ru, xs, 128, 6);
  out_gemm_kernel<<<dim3(2, (BB * NN) / 64), 256, 0, stream>>>(
      xi, xs, xp, Wout_ru, b_ru, gv, 128, 1);
  rhx_kernel<<<(totu + 255) / 256, 256, 0, stream>>>(gv, hx, rh);

  // ---- candidate gconv (O = U = 64), tanh ----
  build_x0_kernel<<<(totx0 + 255) / 256, 256, 0, stream>>>(inputs, rh, x0);
  support_gemm_kernel<<<dim3(FB / 64, NN / 64, 10), 256, 0, stream>>>(
      adj, mask, x0, xk);
  proj_gemm_kernel<<<dim3(1, NN / 64, 11 * BB), 256, 0, stream>>>(
      x0, xk, W_c, xp, 64);
  attention_kernel<<<dim3(NN), 320, 0, stream>>>(xp, Watt_c, Wt1_c, xi, 64, 1);
  attention_kernel<<<dim3(NN), 320, 0, stream>>>(xp, Watt_c, Wt1_c, xs, 64, 6);
  out_gemm_kernel<<<dim3(1, (BB * NN) / 64), 256, 0, stream>>>(
      xi, xs, xp, Wout_c, b_c, cb, 64, 2);

  // ---- GRU combine ----
  final_kernel<<<(totu + 255) / 256, 256, 0, stream>>>(gv, hx, cb, out);
}